// MoEDynamics_20255065768146
// MI455X (gfx1250) — compile-verified
//
#include <hip/hip_runtime.h>
#include <hip/hip_bf16.h>

typedef float v2f __attribute__((ext_vector_type(2)));
typedef float v8f __attribute__((ext_vector_type(8)));
typedef unsigned int u32x4 __attribute__((ext_vector_type(4)));
typedef int i32x8 __attribute__((ext_vector_type(8)));
typedef int i32x4 __attribute__((ext_vector_type(4)));

// ---------------------------------------------------------------------------
// WMMA helper: accumulate a 16x16 f32 tile over Kdim using V_WMMA_F32_16X16X4_F32
// A: row-major 16 x Kdim tile (base = first row), leading dim lda
// B: row-major Kdim x N weight, Bcol pre-offset to column (col0 + r), ldb = N
// Fragment layout per ISA 7.12.2:
//   A 16x4: lane<16 -> M=lane, K={k0,k0+1}; lane>=16 -> M=lane-16, K={k0+2,k0+3}
//   B  4x16: mirrored (row k striped across lanes, K split lo/hi lane halves)
//   C/D: VGPR p: lanes 0-15 -> M=p, lanes 16-31 -> M=p+8; N = lane&15
// ---------------------------------------------------------------------------
__device__ __forceinline__ v8f wmma_acc_tile(const float* __restrict__ Arow, int lda,
                                             const float* __restrict__ Bcol, int ldb,
                                             int Kdim, v8f acc, int r, int kk) {
#pragma unroll
  for (int k0 = 0; k0 < Kdim; k0 += 4) {
    v2f av, bv;
    av.x = Arow[r * lda + k0 + kk];
    av.y = Arow[r * lda + k0 + kk + 1];
    bv.x = Bcol[(k0 + kk) * ldb];
    bv.y = Bcol[(k0 + kk + 1) * ldb];
    acc = __builtin_amdgcn_wmma_f32_16x16x4_f32(false, av, false, bv,
                                                (short)0, acc, false, false);
  }
  return acc;
}

// ---------------------------------------------------------------------------
// Generic GEMM: Y[R x N] = act(X[R x K] @ W[K x N] + bias[N]), R = 512 rows.
// Compile-time K, N -> fully unrolled WMMA chain. 256 threads = 8 waves,
// one 16x16 tile per wave. N may be < 16 (clamped loads, guarded stores).
// ---------------------------------------------------------------------------
template <bool RELU, int K, int N>
__global__ void gemm_kernel(const float* __restrict__ X, const float* __restrict__ W,
                            const float* __restrict__ bias, float* __restrict__ Y) {
  constexpr int tilesN = (N + 15) >> 4;
  int lane = threadIdx.x & 31;
  int wave = threadIdx.x >> 5;
  int tile = blockIdx.x * 8 + wave;
  int tm = tile / tilesN;
  int tn = tile - tm * tilesN;
  int row0 = tm << 4, col0 = tn << 4;

  int r  = lane & 15;
  int kk = (lane >> 4) << 1;   // 0 or 2
  int hi = (lane >> 4) << 3;   // 0 or 8

  int cb = col0 + r;
  if (N < 16) { if (cb > N - 1) cb = N - 1; }      // clamp B column loads
  const float* Arow = X + row0 * K;
  const float* Bcol = W + cb;
  __builtin_prefetch(W, 0, 1);                     // weights reused all 8 steps

  v8f acc = {};
  acc = wmma_acc_tile(Arow, K, Bcol, N, K, acc, r, kk);

  int col = col0 + r;
  if (col < N) {
    float bv = bias[col];
    int rbase = row0 + hi;
#pragma unroll
    for (int p = 0; p < 8; ++p) {
      float v = acc[p] + bv;
      if (RELU) v = fmaxf(v, 0.0f);
      Y[(rbase + p) * N + col] = v;
    }
  }
}

// ---------------------------------------------------------------------------
// Expert kernel: one block per (b, m). h_b staged into LDS via the Tensor
// Data Mover, then:
//   a = h_b @ w1a[m], c = h_b @ w1b[m]                 (WMMA, into LDS)
//   hsum[j,s] = sum_i relu(a[i,s] + c[j,s] + b1[m,s])  (LDS reduction)
//   out[b,m,j,t] = wgt[b,j,m] * ( (hsum @ w2[m])[j,t]/64 + b2[m,t] )
// ---------------------------------------------------------------------------
__global__ void expert_kernel(const float* __restrict__ h, const float* __restrict__ wgt,
                              const float* __restrict__ exp_w1, const float* __restrict__ exp_b1,
                              const float* __restrict__ exp_w2, const float* __restrict__ exp_b2,
                              float* __restrict__ out) {
  // Single shared object -> LDS offset 0 for the TDM descriptor.
  __shared__ float smem[3 * 64 * 128];
  float* sh = smem;               // h_b (reused as hsum)
  float* sa = smem + 64 * 128;    // a
  float* sc = smem + 2 * 64 * 128;// c

  int b = blockIdx.x >> 3;
  int m = blockIdx.x & 7;
  int t = threadIdx.x;            // 256 threads = 8 waves
  int lane = t & 31, wave = t >> 5;
  int r  = lane & 15;
  int kk = (lane >> 4) << 1;
  int hi = (lane >> 4) << 3;

  // ---- TDM: DMA h[b] (64 rows x 128 f32, contiguous) into LDS offset 0 ----
  // D# group 0: count=1 | lds_addr=0 | global_addr[56:0] | type=2
  {
    unsigned long long ga = (unsigned long long)(const void*)(h + b * 64 * 128);
    u32x4 g0 = { 1u,                                   // count=1, no gather
                 0u,                                   // lds_addr = 0
                 (unsigned)ga,                         // global_addr[31:0]
                 (unsigned)((ga >> 32) & 0x01FFFFFFu) | 0x80000000u }; // [56:32] | type=2
    // D# group 1: data_size=4B; tensor 128 x 64, stride 128; tile 128 x 64
    i32x8 g1 = { 0x20000,          // workgroup_mask=0, data_size=2 (4 bytes)
                 128 << 16,        // tensor_dim0 = 128 (bits 79:48)
                 64 << 16,         // tensor_dim1 = 64  (bits 111:80)
                 128 << 16,        // tile_dim0   = 128 (bits 127:112)
                 64,               // tile_dim1   = 64  (bits 143:128), tile_dim2=0
                 128,              // tensor_dim0_stride = 128 (bits 207:160)
                 0, 0 };           // tensor_dim1_stride = 0
    i32x4 gz4 = { 0, 0, 0, 0 };
    i32x8 gz8 = { 0, 0, 0, 0, 0, 0, 0, 0 };
    __builtin_amdgcn_tensor_load_to_lds(g0, g1, gz4, gz4, gz8, 0);
    __builtin_amdgcn_s_wait_tensorcnt(0);
  }
  __syncthreads();

  // a and c GEMMs: 64 tiles total (32 for a, 32 for c), 8 per wave
  const float* w1 = exp_w1 + m * (256 * 128);
  for (int tt = wave; tt < 64; tt += 8) {
    int which = tt >> 5;           // 0 -> a (rows 0..127 of exp_w1[m]), 1 -> c
    int tl = tt & 31;
    int tm = tl >> 3, tn = tl & 7;
    const float* Arow = sh + tm * (16 * 128);
    const float* Bcol = w1 + which * (128 * 128) + tn * 16 + r;
    v8f acc = {};
    acc = wmma_acc_tile(Arow, 128, Bcol, 128, 128, acc, r, kk);
    float* dst = (which ? sc : sa) + (tm * 16 + hi) * 128 + tn * 16 + r;
#pragma unroll
    for (int p = 0; p < 8; ++p) dst[p * 128] = acc[p];
  }
  __syncthreads();

  // hsum[j,s] = sum_i relu(a[i,s] + c[j,s] + b1[s]); write into sh (h dead)
  {
    const float* b1 = exp_b1 + m * 128;
    int j  = t >> 2;
    int s0 = t & 3;
    for (int s = s0; s < 128; s += 4) {
      float cj = sc[j * 128 + s] + b1[s];
      float accv = 0.0f;
      for (int i = 0; i < 64; ++i) accv += fmaxf(sa[i * 128 + s] + cj, 0.0f);
      sh[j * 128 + s] = accv;
    }
  }
  __syncthreads();

  // inter = hsum @ w2[m]; fold router weight, 1/K and b2; 32 tiles, 4/wave
  const float* w2 = exp_w2 + m * (128 * 128);
  const float* b2 = exp_b2 + m * 128;
  float* ob = out + (b * 8 + m) * (64 * 128);
  for (int tl = wave; tl < 32; tl += 8) {
    int tm = tl >> 3, tn = tl & 7;
    const float* Arow = sh + tm * (16 * 128);
    const float* Bcol = w2 + tn * 16 + r;
    v8f acc = {};
    acc = wmma_acc_tile(Arow, 128, Bcol, 128, 128, acc, r, kk);
    int col = tn * 16 + r;
    int j0 = tm * 16 + hi;
#pragma unroll
    for (int p = 0; p < 8; ++p) {
      int j = j0 + p;
      float w = wgt[(b * 64 + j) * 8 + m];
      ob[j * 128 + col] = w * (acc[p] * (1.0f / 64.0f) + b2[col]);
    }
  }
}

// ---------------------------------------------------------------------------
// Small elementwise kernels
// ---------------------------------------------------------------------------
__global__ void init_kernel(const float* __restrict__ gt, float* __restrict__ x0,
                            float* __restrict__ out) {
  int idx = blockIdx.x * blockDim.x + threadIdx.x;
  if (idx < 4096) {
    // x0[row, d] = gt[b, 0, k, d]; row = b*64 + k
    int row = idx >> 3, d = idx & 7;
    int b = row >> 6, k = row & 63;
    x0[idx] = gt[((b * 9 + 0) * 64 + k) * 8 + d];
  } else if (idx < 4096 + 32768) {
    // target_states[b, t, k, d] = gt[b, t+1, k, d]
    int j = idx - 4096;
    int d = j & 7, k = (j >> 3) & 63, tt = (j >> 9) & 7, b = j >> 12;
    out[32768 + j] = gt[((b * 9 + tt + 1) * 64 + k) * 8 + d];
  }
}

__global__ void softmax8_kernel(const float* __restrict__ logits, float* __restrict__ w) {
  int row = blockIdx.x * blockDim.x + threadIdx.x;
  if (row < 512) {
    const float* l = logits + row * 8;
    float mx = l[0];
#pragma unroll
    for (int i = 1; i < 8; ++i) mx = fmaxf(mx, l[i]);
    float e[8], s = 0.0f;
#pragma unroll
    for (int i = 0; i < 8; ++i) { e[i] = __expf(l[i] - mx); s += e[i]; }
    float inv = 1.0f / s;
#pragma unroll
    for (int i = 0; i < 8; ++i) w[row * 8 + i] = e[i] * inv;
  }
}

__global__ void layernorm_kernel(const float* __restrict__ x, const float* __restrict__ g,
                                 const float* __restrict__ bt, float* __restrict__ y, int L) {
  int wave = threadIdx.x >> 5, lane = threadIdx.x & 31;
  int row = blockIdx.x * (blockDim.x >> 5) + wave;
  const float* xr = x + row * L;
  float* yr = y + row * L;
  float s = 0.0f, s2 = 0.0f;
  for (int i = lane; i < L; i += 32) { float v = xr[i]; s += v; s2 += v * v; }
#pragma unroll
  for (int o = 16; o > 0; o >>= 1) {
    s  += __shfl_xor(s, o, 32);
    s2 += __shfl_xor(s2, o, 32);
  }
  float invL = 1.0f / (float)L;
  float mu = s * invL;
  float var = s2 * invL - mu * mu;
  float rs = rsqrtf(var + 1e-5f);
  for (int i = lane; i < L; i += 32) yr[i] = (xr[i] - mu) * rs * g[i] + bt[i];
}

// combine experts over m (deterministic ordered sum) + build concat buffer
__global__ void combine_kernel(const float* __restrict__ expert, const float* __restrict__ ds,
                               float* __restrict__ cat) {
  int idx = blockIdx.x * blockDim.x + threadIdx.x;   // 512*128
  int row = idx >> 7, tcol = idx & 127;
  int b = row >> 6, j = row & 63;
  float s = 0.0f;
#pragma unroll
  for (int m = 0; m < 8; ++m) s += expert[((b * 8 + m) * 64 + j) * 128 + tcol];
  cat[row * 256 + tcol]       = ds[row * 128 + tcol];
  cat[row * 256 + 128 + tcol] = s;
}

__global__ void copy_pred_kernel(const float* __restrict__ pred, float* __restrict__ out,
                                 int step) {
  int idx = blockIdx.x * blockDim.x + threadIdx.x;   // 4096
  int row = idx >> 3, d = idx & 7;
  int b = row >> 6, k = row & 63;
  out[b * 4096 + step * 512 + k * 8 + d] = pred[idx];
}

// ---------------------------------------------------------------------------
// Host side
// ---------------------------------------------------------------------------
template <bool RELU, int K, int N>
static inline void launch_gemm(const float* X, const float* W, const float* bias,
                               float* Y, hipStream_t s) {
  constexpr int tilesN = (N + 15) / 16;
  constexpr int blocks = 32 * tilesN / 8;   // R=512 -> 32 row tiles, 8 waves/block
  gemm_kernel<RELU, K, N><<<blocks, 256, 0, s>>>(X, W, bias, Y);
}

extern "C" void kernel_launch(void* const* d_in, const int* in_sizes, int n_in,
                              void* d_out, int out_size, void* d_ws, size_t ws_size,
                              hipStream_t stream) {
  (void)in_sizes; (void)n_in; (void)out_size; (void)ws_size;

  const float* gt      = (const float*)d_in[0];
  const float* enc_w1  = (const float*)d_in[1];
  const float* enc_b1  = (const float*)d_in[2];
  const float* enc_w2  = (const float*)d_in[3];
  const float* enc_b2  = (const float*)d_in[4];
  const float* rout_w1 = (const float*)d_in[5];
  const float* rout_b1 = (const float*)d_in[6];
  const float* rout_w2 = (const float*)d_in[7];
  const float* rout_b2 = (const float*)d_in[8];
  const float* fs_g    = (const float*)d_in[9];
  const float* fs_bt   = (const float*)d_in[10];
  const float* fs_w1   = (const float*)d_in[11];
  const float* fs_b1   = (const float*)d_in[12];
  const float* fs_w2   = (const float*)d_in[13];
  const float* fs_b2   = (const float*)d_in[14];
  const float* exp_w1  = (const float*)d_in[15];
  const float* exp_b1  = (const float*)d_in[16];
  const float* exp_w2  = (const float*)d_in[17];
  const float* exp_b2  = (const float*)d_in[18];
  const float* up_g    = (const float*)d_in[19];
  const float* up_bt   = (const float*)d_in[20];
  const float* up_w1   = (const float*)d_in[21];
  const float* up_b1   = (const float*)d_in[22];
  const float* up_w2   = (const float*)d_in[23];
  const float* up_b2   = (const float*)d_in[24];
  const float* dec_w1  = (const float*)d_in[25];
  const float* dec_b1  = (const float*)d_in[26];
  const float* dec_w2  = (const float*)d_in[27];
  const float* dec_b2  = (const float*)d_in[28];
  // d_in[29] = rollout_steps = 8 (fixed by harness setup; T-1 == 8 bounds it)

  float* ws = (float*)d_ws;
  float* out = (float*)d_out;

  float* h      = ws;                 // 512*128
  float* tmp    = ws + 65536;         // 512*128
  float* sm     = ws + 131072;        // 512*8 (x0 at init, pred per step)
  float* logits = ws + 135168;        // 512*8
  float* wgt    = ws + 139264;        // 512*8
  float* lnh    = ws + 143360;        // 512*128
  float* ds     = ws + 208896;        // 512*128
  float* cat    = ws + 274432;        // 512*256
  float* lncat  = ws + 405504;        // 512*256
  float* expert = ws + 536576;        // 8*8*64*128

  // gather x0 = gt[:,0] and write target_states = gt[:,1:9]
  init_kernel<<<144, 256, 0, stream>>>(gt, sm, out);

  // h = mlp2(x0, enc)
  launch_gemm<true, 8, 128>(sm, enc_w1, enc_b1, tmp, stream);
  launch_gemm<false, 128, 128>(tmp, enc_w2, enc_b2, h, stream);

  for (int step = 0; step < 8; ++step) {
    // router weights
    launch_gemm<true, 128, 128>(h, rout_w1, rout_b1, tmp, stream);
    launch_gemm<false, 128, 8>(tmp, rout_w2, rout_b2, logits, stream);
    softmax8_kernel<<<2, 256, 0, stream>>>(logits, wgt);
    // delta_self = mlp2(ln(h), fs)
    layernorm_kernel<<<64, 256, 0, stream>>>(h, fs_g, fs_bt, lnh, 128);
    launch_gemm<true, 128, 128>(lnh, fs_w1, fs_b1, tmp, stream);
    launch_gemm<false, 128, 128>(tmp, fs_w2, fs_b2, ds, stream);
    // expert interaction term (factored over the pairwise i-sum)
    expert_kernel<<<64, 256, 0, stream>>>(h, wgt, exp_w1, exp_b1, exp_w2, exp_b2, expert);
    combine_kernel<<<256, 256, 0, stream>>>(expert, ds, cat);
    // h = mlp2(ln(cat), up)
    layernorm_kernel<<<64, 256, 0, stream>>>(cat, up_g, up_bt, lncat, 256);
    launch_gemm<true, 256, 128>(lncat, up_w1, up_b1, tmp, stream);
    launch_gemm<false, 128, 128>(tmp, up_w2, up_b2, h, stream);
    // pred = mlp2(h, dec); store; h = mlp2(pred, enc)
    launch_gemm<true, 128, 128>(h, dec_w1, dec_b1, tmp, stream);
    launch_gemm<false, 128, 8>(tmp, dec_w2, dec_b2, sm, stream);
    copy_pred_kernel<<<16, 256, 0, stream>>>(sm, out, step);
    launch_gemm<true, 8, 128>(sm, enc_w1, enc_b1, tmp, stream);
    launch_gemm<false, 128, 128>(tmp, enc_w2, enc_b2, h, stream);
  }
}